// PWM_49168785604887
// MI455X (gfx1250) — compile-verified
//
#include <hip/hip_runtime.h>
#include <hip/hip_bf16.h>

// Problem constants (from reference): B=256 batches, GROUPS=256, 4 channels,
// K=19 taps, L=1000, conv output length T = L-K+1 = 982.
// out[b,g] = sum_cc pwms[g,cc] * (S[b,cc] + S[b,75-cc]) + 2*T*bias[g]
// where S[b, c*19+k] = sum_{l=k}^{k+981} x[b,c,l].

typedef float v2f __attribute__((ext_vector_type(2)));
typedef float v8f __attribute__((ext_vector_type(8)));

// ---------------------------------------------------------------------------
// Kernel 1: per (b,c) row of 1000 floats, compute the 19 windowed sums
//   S[row*19 + k] = Total - head_excl(k) - tail(18-k)
// One wave32 per row; 8 waves (256 threads) per block; 1024 rows total.
// This kernel is the roofline-dominant part: 4.1 MB of x streamed once
// (~0.19 us at 23.3 TB/s).
// ---------------------------------------------------------------------------
__global__ __launch_bounds__(256) void pwm_row_windows(
    const float* __restrict__ x, float* __restrict__ S)
{
    const int lane = threadIdx.x & 31;
    const int wave = threadIdx.x >> 5;
    const int row  = blockIdx.x * 8 + wave;          // (b*4 + c), 0..1023
    const float* __restrict__ xr = x + row * 1000;

    // Full-row sum: coalesced strided loads, then wave32 butterfly reduce.
    float total = 0.0f;
    for (int l = lane; l < 1000; l += 32) total += xr[l];
#pragma unroll
    for (int off = 16; off > 0; off >>= 1)
        total += __shfl_xor(total, off, 32);

    // First 19 and last 19 elements -> inclusive wave scans.
    float h = (lane < 19) ? xr[lane]        : 0.0f;  // head element x[lane]
    float t = (lane < 19) ? xr[999 - lane]  : 0.0f;  // tail element x[999-lane]
    float hI = h, tI = t;
#pragma unroll
    for (int off = 1; off < 32; off <<= 1) {
        float ho = __shfl_up(hI, off, 32);
        float to = __shfl_up(tI, off, 32);
        if (lane >= off) { hI += ho; tI += to; }
    }
    // head_excl(k) = sum of first k elems = inclusive(k) - x[k]
    const float head_excl = hI - h;
    // tail(m) = sum of last m elems = tI at lane m-1; need m = 18-lane.
    const int   src     = (lane < 18) ? (17 - lane) : 0;
    const float twv     = __shfl(tI, src, 32);
    const float tailwin = (lane < 18) ? twv : 0.0f;

    if (lane < 19)
        S[row * 19 + lane] = total - head_excl - tailwin;
}

// ---------------------------------------------------------------------------
// Kernel 2: 256x256 f32 GEMM via V_WMMA_F32_16X16X4_F32.
//   out[b,g] = sum_{cc<76} W[g,cc] * (S[b,cc] + S[b,75-cc]) + 1964*bias[g]
// One wave32 per 16x16 tile; 4 waves per block; 64 blocks = 256 tiles.
// Fully unrolled; all three data streams are aligned 8-byte loads from fixed
// base pointers with constant (ascending / descending) offsets, so every
// global_load_b64 uses an immediate IOFFSET and clauses cleanly; the 19 WMMAs
// chain only through the C accumulator.
// A-frag (16x4 f32, 2 VGPRs): lanes 0-15 hold {K=0,K=1}, lanes 16-31 {K=2,K=3}.
// B-frag mirrored; C/D: VGPR r -> M=r (lanes 0-15) / M=8+r (lanes 16-31).
// ---------------------------------------------------------------------------
__global__ __launch_bounds__(128) void pwm_gemm_wmma(
    const float* __restrict__ S,     // [256][76]  (flattened 1024*19)
    const float* __restrict__ W,     // pwms: [256][76]
    const float* __restrict__ bias,  // [256]
    float* __restrict__ out)         // [256][256] row-major (b, g)
{
    const int lane = threadIdx.x & 31;
    const int tile = blockIdx.x * 4 + (threadIdx.x >> 5); // 0..255
    const int m0 = (tile >> 4) << 4;                      // batch-tile base
    const int n0 = (tile & 15) << 4;                      // group-tile base

    const int half = lane >> 4;     // 0: K pair {0,1}; 1: K pair {2,3}
    const int l16  = lane & 15;
    const int m = m0 + l16;         // A-matrix row for this lane
    const int n = n0 + l16;         // B-matrix col for this lane

    // Base pointers (row bases are 304 B = 16B-aligned; all pair offsets even
    // -> 8B-aligned v2f loads).
    const float* __restrict__ Sf = S + m * 76 + half * 2;        // forward
    const float* __restrict__ Sr = S + m * 76 + 74 - half * 2;   // reversed
    const float* __restrict__ Wf = W + n * 76 + half * 2;        // weights

    v8f acc = {};
#pragma unroll
    for (int kb = 0; kb < 19; ++kb) {
        const v2f fw = *(const v2f*)(Sf + 4 * kb);   // {S[k0],    S[k0+1]}
        const v2f rv = *(const v2f*)(Sr - 4 * kb);   // {S[74-k0], S[75-k0]}
        const v2f b  = *(const v2f*)(Wf + 4 * kb);   // {W[k0],    W[k0+1]}
        v2f a;
        a.x = fw.x + rv.y;   // S[k0]   + S[75-k0]
        a.y = fw.y + rv.x;   // S[k0+1] + S[74-k0]
        // 8 args: (neg_a, A, neg_b, B, c_mod, C, reuse_a, reuse_b)
        acc = __builtin_amdgcn_wmma_f32_16x16x4_f32(
            false, a, false, b, (short)0, acc, false, false);
    }

    // Epilogue: bias term = 2*T*bias[g] with T=982.
    const int   mrow  = m0 + half * 8;
    const float bterm = 1964.0f * bias[n];
#pragma unroll
    for (int r = 0; r < 8; ++r)
        out[(mrow + r) * 256 + n] = acc[r] + bterm;
}

// ---------------------------------------------------------------------------
extern "C" void kernel_launch(void* const* d_in, const int* in_sizes, int n_in,
                              void* d_out, int out_size, void* d_ws, size_t ws_size,
                              hipStream_t stream)
{
    const float* x    = (const float*)d_in[0];   // (256, 4, 1000)
    const float* pwms = (const float*)d_in[1];   // (256, 4, 19)
    const float* bias = (const float*)d_in[2];   // (256,)
    float* out = (float*)d_out;                  // (256, 256)

    float* S = (float*)d_ws;                     // 1024*19 floats = 77824 B

    // 1024 rows, 8 waves/block -> 128 blocks of 256 threads.
    pwm_row_windows<<<128, 256, 0, stream>>>(x, S);
    // 256 tiles, 4 waves/block -> 64 blocks of 128 threads.
    pwm_gemm_wmma<<<64, 128, 0, stream>>>(S, pwms, bias, out);
}